// AttnDecoderRNN_6141803233582
// MI455X (gfx1250) — compile-verified
//
#include <hip/hip_runtime.h>
#include <hip/hip_bf16.h>

#define H 1024
#define V 50257
#define L 64

typedef __attribute__((ext_vector_type(2))) float v2f;
typedef __attribute__((ext_vector_type(8))) float v8f;

// ---------------------------------------------------------------------------
// Generic fp32 GEMV via V_WMMA_F32_16X16X4_F32.
// y[r] = dot(W[r, 0..K), x) + bias[r], optional ReLU.
// One wave per 16-row tile. A = 16x4 tile of W, B = x broadcast to 16 cols.
// blockDim.x = 128 (4 waves).
// NT=true streams W with non-temporal loads (read-once data > L2 capacity),
// preserving L2 residency of the small reused weight matrices across replays.
// ---------------------------------------------------------------------------
template <bool NT>
__global__ void gemv16_wmma_kernel(const float* __restrict__ W,
                                   const float* __restrict__ x,
                                   const float* __restrict__ bias,
                                   float* __restrict__ y,
                                   int rows, int K, int do_relu) {
    const int lane = threadIdx.x & 31;
    const int wave = threadIdx.x >> 5;
    const int ntiles = (rows + 15) >> 4;
    const int tile = blockIdx.x * (blockDim.x >> 5) + wave;
    if (tile >= ntiles) return;                 // wave-uniform exit

    const int row0 = tile << 4;
    int row = row0 + (lane & 15);
    if (row >= rows) row = rows - 1;            // branchless clamp (loads only)
    const int koff = (lane >> 4) << 1;          // 0 for lanes 0-15, 2 for 16-31

    const float* wp = W + (size_t)row * (size_t)K + koff;
    const float* xp = x + koff;

    v8f acc = {};
#pragma unroll 8
    for (int k = 0; k < K; k += 4) {
        v2f a;
        if (NT) a = __builtin_nontemporal_load((const v2f*)(wp + k));
        else    a = *(const v2f*)(wp + k);      // A: W[row][k+koff .. k+koff+1]
        v2f b = *(const v2f*)(xp + k);          // B: x[k+koff .. k+koff+1] (all 16 cols)
        acc = __builtin_amdgcn_wmma_f32_16x16x4_f32(
            /*neg_a=*/false, a, /*neg_b=*/false, b,
            /*c_mod=*/(short)0, acc, /*reuse_a=*/false, /*reuse_b=*/false);
    }

    // Column N=0 lives in lane 0 (rows row0..row0+7) and lane 16 (rows +8..+15).
    if (lane == 0 || lane == 16) {
        const int base = row0 + (lane >> 1);    // lane16 -> +8
#pragma unroll
        for (int i = 0; i < 8; ++i) {
            const int rr = base + i;
            if (rr < rows) {
                float v = acc[i] + bias[rr];
                if (do_relu) v = fmaxf(v, 0.0f);
                y[rr] = v;
            }
        }
    }
}

// ---------------------------------------------------------------------------
// cat3 = [emb[input], h0, h1]; comb[0..H) = embedded
// ---------------------------------------------------------------------------
__global__ void prep_cat3_kernel(const int* __restrict__ input,
                                 const float* __restrict__ emb,
                                 const float* __restrict__ hidden,
                                 float* __restrict__ cat3,
                                 float* __restrict__ comb) {
    const int i = blockIdx.x * blockDim.x + threadIdx.x;
    if (i >= 3 * H) return;
    if (i < H) {
        float e = emb[(size_t)input[0] * H + i];
        cat3[i] = e;
        comb[i] = e;
    } else {
        cat3[i] = hidden[i - H];   // covers h0 then h1 contiguously
    }
}

// ---------------------------------------------------------------------------
// softmax over 64 logits; attn_applied = w @ enc_outputs -> comb[H..3H);
// also emit attn_weights to d_out. Single block of 256.
// ---------------------------------------------------------------------------
__global__ void attn_softmax_kernel(const float* __restrict__ logits,
                                    const float* __restrict__ enc,
                                    float* __restrict__ comb,
                                    float* __restrict__ attnw_out) {
    __shared__ float sw[L];
    const int tid = threadIdx.x;
    if (tid == 0) {
        float m = logits[0];
        for (int l = 1; l < L; ++l) m = fmaxf(m, logits[l]);
        float s = 0.0f;
        for (int l = 0; l < L; ++l) { float e = __expf(logits[l] - m); sw[l] = e; s += e; }
        float inv = 1.0f / s;
        for (int l = 0; l < L; ++l) sw[l] *= inv;
    }
    __syncthreads();
    if (tid < L) attnw_out[tid] = sw[tid];
    for (int j = tid; j < 2 * H; j += blockDim.x) {
        float acc = 0.0f;
#pragma unroll 8
        for (int l = 0; l < L; ++l) acc += sw[l] * enc[l * 2 * H + j];
        comb[H + j] = acc;
    }
}

// ---------------------------------------------------------------------------
// GRU gate combine (both directions), emit gru_out and new_hidden.
// ---------------------------------------------------------------------------
__device__ __forceinline__ float sigm(float v) { return 1.0f / (1.0f + __expf(-v)); }

__global__ void gru_combine_kernel(const float* __restrict__ gif, const float* __restrict__ ghf,
                                   const float* __restrict__ gib, const float* __restrict__ ghb,
                                   const float* __restrict__ hidden,
                                   float* __restrict__ gruo, float* __restrict__ hid_out) {
    const int i = blockIdx.x * blockDim.x + threadIdx.x;
    if (i >= H) return;
    // forward
    float rf = sigm(gif[i] + ghf[i]);
    float zf = sigm(gif[H + i] + ghf[H + i]);
    float nf = tanhf(gif[2 * H + i] + rf * ghf[2 * H + i]);
    float hf = (1.0f - zf) * nf + zf * hidden[i];
    // backward
    float rb = sigm(gib[i] + ghb[i]);
    float zb = sigm(gib[H + i] + ghb[H + i]);
    float nb = tanhf(gib[2 * H + i] + rb * ghb[2 * H + i]);
    float hb = (1.0f - zb) * nb + zb * hidden[H + i];

    gruo[i] = hf;        gruo[H + i] = hb;
    hid_out[i] = hf;     hid_out[H + i] = hb;
}

// ---------------------------------------------------------------------------
// Deterministic log-softmax over V: two-level max, two-level sum(exp), write.
// ---------------------------------------------------------------------------
__global__ void reduce_max_kernel(const float* __restrict__ v, float* __restrict__ pmax, int n) {
    __shared__ float sm[256];
    const int tid = threadIdx.x;
    float m = -3.402823466e38f;
    for (int i = blockIdx.x * blockDim.x + tid; i < n; i += gridDim.x * blockDim.x)
        m = fmaxf(m, v[i]);
    sm[tid] = m; __syncthreads();
    for (int s = 128; s > 0; s >>= 1) {
        if (tid < s) sm[tid] = fmaxf(sm[tid], sm[tid + s]);
        __syncthreads();
    }
    if (tid == 0) pmax[blockIdx.x] = sm[0];
}

__global__ void final_max_kernel(const float* __restrict__ pmax, float* __restrict__ gmax) {
    __shared__ float sm[64];
    const int tid = threadIdx.x;
    sm[tid] = pmax[tid]; __syncthreads();
    for (int s = 32; s > 0; s >>= 1) {
        if (tid < s) sm[tid] = fmaxf(sm[tid], sm[tid + s]);
        __syncthreads();
    }
    if (tid == 0) *gmax = sm[0];
}

__global__ void reduce_sumexp_kernel(const float* __restrict__ v, const float* __restrict__ gmax,
                                     float* __restrict__ psum, int n) {
    __shared__ float sm[256];
    const int tid = threadIdx.x;
    const float gm = *gmax;
    float s = 0.0f;
    for (int i = blockIdx.x * blockDim.x + tid; i < n; i += gridDim.x * blockDim.x)
        s += __expf(v[i] - gm);
    sm[tid] = s; __syncthreads();
    for (int st = 128; st > 0; st >>= 1) {
        if (tid < st) sm[tid] += sm[tid + st];
        __syncthreads();
    }
    if (tid == 0) psum[blockIdx.x] = sm[0];
}

__global__ void final_lse_kernel(const float* __restrict__ psum, const float* __restrict__ gmax,
                                 float* __restrict__ lse) {
    __shared__ float sm[64];
    const int tid = threadIdx.x;
    sm[tid] = psum[tid]; __syncthreads();
    for (int s = 32; s > 0; s >>= 1) {
        if (tid < s) sm[tid] += sm[tid + s];
        __syncthreads();
    }
    if (tid == 0) *lse = *gmax + __logf(sm[0]);
}

__global__ void write_logp_kernel(const float* __restrict__ logits, const float* __restrict__ lse,
                                  float* __restrict__ out, int n) {
    const int i = blockIdx.x * blockDim.x + threadIdx.x;
    if (i < n) out[i] = logits[i] - *lse;
}

// ---------------------------------------------------------------------------
extern "C" void kernel_launch(void* const* d_in, const int* in_sizes, int n_in,
                              void* d_out, int out_size, void* d_ws, size_t ws_size,
                              hipStream_t stream) {
    (void)in_sizes; (void)n_in; (void)out_size; (void)ws_size;

    const int*   input  = (const int*)  d_in[0];
    const float* hidden = (const float*)d_in[1];   // (2,1,H)
    // d_in[2] = encoder_output  (unused by reference)
    const float* enc    = (const float*)d_in[3];   // (L, 2H)
    const float* emb    = (const float*)d_in[4];   // (V, H)
    const float* attn_W = (const float*)d_in[5];   // (L, 3H)
    const float* attn_b = (const float*)d_in[6];
    const float* comb_W = (const float*)d_in[7];   // (H, 3H)
    const float* comb_b = (const float*)d_in[8];
    const float* Wih_f  = (const float*)d_in[9];   // (3H, H)
    const float* Whh_f  = (const float*)d_in[10];
    const float* bih_f  = (const float*)d_in[11];
    const float* bhh_f  = (const float*)d_in[12];
    const float* Wih_b  = (const float*)d_in[13];
    const float* Whh_b  = (const float*)d_in[14];
    const float* bih_b  = (const float*)d_in[15];
    const float* bhh_b  = (const float*)d_in[16];
    const float* out_W  = (const float*)d_in[17];  // (V, 2H)
    const float* out_b  = (const float*)d_in[18];

    float* out_logp  = (float*)d_out;          // V
    float* out_hid   = out_logp + V;           // 2H
    float* out_attnw = out_logp + V + 2 * H;   // L

    float* ws     = (float*)d_ws;
    float* cat3   = ws;              // 3072
    float* comb   = ws + 3072;       // 3072
    float* attnl  = ws + 6144;       // 64
    float* xbuf   = ws + 6208;       // 1024
    float* gif    = ws + 7232;       // 3072
    float* ghf    = ws + 10304;      // 3072
    float* gib    = ws + 13376;      // 3072
    float* ghb    = ws + 16448;      // 3072
    float* gruo   = ws + 19520;      // 2048
    float* logits = ws + 21568;      // 50257
    float* pmax   = ws + 71825;      // 64
    float* psum   = ws + 71889;      // 64
    float* gmax   = ws + 71953;      // 1
    float* lse    = ws + 71954;      // 1

    auto gemv_grid = [](int rows) { int nt = (rows + 15) >> 4; return (nt + 3) >> 2; };

    // 1. embedded gather + cat3 + comb[0..H)
    prep_cat3_kernel<<<(3 * H + 255) / 256, 256, 0, stream>>>(input, emb, hidden, cat3, comb);

    // 2. attn logits = cat3 @ attn_W.T + attn_b
    gemv16_wmma_kernel<false><<<gemv_grid(L), 128, 0, stream>>>(attn_W, cat3, attn_b, attnl, L, 3 * H, 0);

    // 3. softmax(64) + attn_applied -> comb[H..3H), attn_weights -> d_out
    attn_softmax_kernel<<<1, 256, 0, stream>>>(attnl, enc, comb, out_attnw);

    // 4. x = relu(comb @ comb_W.T + comb_b)
    gemv16_wmma_kernel<false><<<gemv_grid(H), 128, 0, stream>>>(comb_W, comb, comb_b, xbuf, H, 3 * H, 1);

    // 5. GRU gate GEMVs (both directions) — small, L2-resident weights: RT hints
    gemv16_wmma_kernel<false><<<gemv_grid(3 * H), 128, 0, stream>>>(Wih_f, xbuf,       bih_f, gif, 3 * H, H, 0);
    gemv16_wmma_kernel<false><<<gemv_grid(3 * H), 128, 0, stream>>>(Whh_f, hidden,     bhh_f, ghf, 3 * H, H, 0);
    gemv16_wmma_kernel<false><<<gemv_grid(3 * H), 128, 0, stream>>>(Wih_b, xbuf,       bih_b, gib, 3 * H, H, 0);
    gemv16_wmma_kernel<false><<<gemv_grid(3 * H), 128, 0, stream>>>(Whh_b, hidden + H, bhh_b, ghb, 3 * H, H, 0);

    // 6. gate nonlinearity -> gru_out, new_hidden
    gru_combine_kernel<<<(H + 255) / 256, 256, 0, stream>>>(gif, ghf, gib, ghb, hidden, gruo, out_hid);

    // 7. vocab logits = gru_out @ out_W.T + out_b
    //    412 MB read-once stream > 192 MB L2: non-temporal so it doesn't evict
    //    the resident small weights between graph replays.
    gemv16_wmma_kernel<true><<<gemv_grid(V), 128, 0, stream>>>(out_W, gruo, out_b, logits, V, 2 * H, 0);

    // 8. log_softmax over V (deterministic two-level reductions)
    reduce_max_kernel   <<<64, 256, 0, stream>>>(logits, pmax, V);
    final_max_kernel    <<<1,  64,  0, stream>>>(pmax, gmax);
    reduce_sumexp_kernel<<<64, 256, 0, stream>>>(logits, gmax, psum, V);
    final_lse_kernel    <<<1,  64,  0, stream>>>(psum, gmax, lse);
    write_logp_kernel   <<<(V + 255) / 256, 256, 0, stream>>>(logits, lse, out_logp, V);
}